// HybridLayer_29970281791921
// MI455X (gfx1250) — compile-verified
//
#include <hip/hip_runtime.h>

#define B_   4
#define N_   4096
#define CIN  64
#define COUT 128
#define KE   20
#define KP   16
#define NH   4
#define DPH  32

typedef __attribute__((ext_vector_type(2))) float v2f;
typedef __attribute__((ext_vector_type(8))) float v8f;

__device__ __forceinline__ v8f wmma_f32_4(v2f a, v2f b, v8f c) {
  // D(16x16,f32) += A(16x4,f32) * B(4x16,f32); exact f32 math (matches reference dtype)
  return __builtin_amdgcn_wmma_f32_16x16x4_f32(false, a, false, b, (short)0, c, false, false);
}

#define V8F_ZERO {0.f,0.f,0.f,0.f,0.f,0.f,0.f,0.f}

// Branchless stable bubble-insert into a descending sorted list of length K.
// Strict '>' preserves jax.lax.top_k tie-break (lower index first).
template <int K>
__device__ __forceinline__ void topk_insert(float (&kv)[K], int (&ki)[K], float v, int vi) {
  if (v > kv[K - 1]) {
    float cv = v; int ci = vi;
#pragma unroll
    for (int q = 0; q < K; ++q) {
      bool gt = cv > kv[q];
      float nv = gt ? cv : kv[q];
      float dv = gt ? kv[q] : cv;
      int   ni = gt ? ci : ki[q];
      int   di = gt ? ki[q] : ci;
      kv[q] = nv; ki[q] = ni; cv = dv; ci = di;
    }
  }
}

// ---------------------------------------------------------------- sqnorms ---
__global__ void sqnorm_kernel(const float* __restrict__ x, const float* __restrict__ pos,
                              float* __restrict__ sqx, float* __restrict__ sqp) {
  int p = blockIdx.x * blockDim.x + threadIdx.x;
  if (p >= B_ * N_) return;
  const float* xr = x + (size_t)p * CIN;
  float s = 0.f;
#pragma unroll
  for (int c = 0; c < CIN; ++c) s += xr[c] * xr[c];
  sqx[p] = s;
  const float* pr = pos + (size_t)p * 3;
  sqp[p] = pr[0]*pr[0] + pr[1]*pr[1] + pr[2]*pr[2];
}

// ------------------------------------------------------------- KNN on pos ---
// One wave = 16 query points. Gram tiles via one f32 WMMA (K=3, padded to 4).
__global__ __launch_bounds__(32) void knn_pos_kernel(const float* __restrict__ pos,
                                                     const float* __restrict__ sqp,
                                                     int* __restrict__ idx) {
  __shared__ float sqq[16];
  __shared__ float nd[16][17];
  int wid  = blockIdx.x;
  int b    = wid / (N_ / 16);
  int m0   = (wid % (N_ / 16)) * 16;
  int lane = threadIdx.x;
  int half = lane >> 4, mr = lane & 15;
  const float* pb = pos + (size_t)b * N_ * 3;
  const float* sb = sqp + (size_t)b * N_;

  v2f a;
  {
    const float* qr = pb + (size_t)(m0 + mr) * 3;
    a.x = (half == 0) ? qr[0] : qr[2];
    a.y = (half == 0) ? qr[1] : 0.f;
  }
  if (lane < 16) sqq[lane] = sb[m0 + lane];
  __syncthreads();

  float kv[KP]; int ki[KP];
#pragma unroll
  for (int j = 0; j < KP; ++j) { kv[j] = -3.0e38f; ki[j] = 0; }

  for (int t = 0; t < N_ / 16; ++t) {
    int n0 = t * 16;
    v2f bb;
    {
      const float* cr = pb + (size_t)(n0 + mr) * 3;
      bb.x = (half == 0) ? cr[0] : cr[2];
      bb.y = (half == 0) ? cr[1] : 0.f;
    }
    v8f acc = V8F_ZERO;
    acc = wmma_f32_4(a, bb, acc);
    float sqn = sb[n0 + mr];
#pragma unroll
    for (int r = 0; r < 8; ++r) {
      int m = r + 8 * half;
      nd[m][mr] = 2.f * acc[r] - sqq[m] - sqn;
    }
    __syncthreads();
    if (lane < 16) {
#pragma unroll
      for (int j2 = 0; j2 < 16; ++j2) topk_insert<KP>(kv, ki, nd[lane][j2], n0 + j2);
    }
    __syncthreads();
  }
  if (lane < 16) {
    int* op = idx + ((size_t)b * N_ + m0 + lane) * KP;
#pragma unroll
    for (int j = 0; j < KP; ++j) op[j] = ki[j];
  }
}

// -------------------------------------------------------- KNN on features ---
// One wave = 16 query points; candidate Gram tiles: 16 WMMAs (K=64) split over
// two independent accumulator chains for ILP.
__global__ __launch_bounds__(32) void knn_feat_kernel(const float* __restrict__ x,
                                                      const float* __restrict__ sqx,
                                                      int* __restrict__ idx) {
  __shared__ float sqq[16];
  __shared__ float nd[16][17];
  int wid  = blockIdx.x;
  int b    = wid / (N_ / 16);
  int m0   = (wid % (N_ / 16)) * 16;
  int lane = threadIdx.x;
  int half = lane >> 4, mr = lane & 15;
  const float* xb = x + (size_t)b * N_ * CIN;
  const float* sb = sqx + (size_t)b * N_;

  v2f A[16];
  {
    const float2* qr = (const float2*)(xb + (size_t)(m0 + mr) * CIN);
#pragma unroll
    for (int kk = 0; kk < 16; ++kk) { float2 t2 = qr[2 * kk + half]; A[kk].x = t2.x; A[kk].y = t2.y; }
  }
  if (lane < 16) sqq[lane] = sb[m0 + lane];
  __syncthreads();

  float kv[KE]; int ki[KE];
#pragma unroll
  for (int j = 0; j < KE; ++j) { kv[j] = -3.0e38f; ki[j] = 0; }

  for (int t = 0; t < N_ / 16; ++t) {
    int n0 = t * 16;
    const float2* cr = (const float2*)(xb + (size_t)(n0 + mr) * CIN);
    v8f acc0 = V8F_ZERO, acc1 = V8F_ZERO;
#pragma unroll
    for (int kk = 0; kk < 16; kk += 2) {
      float2 t0 = cr[2 * kk + half];
      float2 t1 = cr[2 * (kk + 1) + half];
      v2f b0; b0.x = t0.x; b0.y = t0.y;
      v2f b1; b1.x = t1.x; b1.y = t1.y;
      acc0 = wmma_f32_4(A[kk], b0, acc0);
      acc1 = wmma_f32_4(A[kk + 1], b1, acc1);
    }
    float sqn = sb[n0 + mr];
#pragma unroll
    for (int r = 0; r < 8; ++r) {
      int m = r + 8 * half;
      nd[m][mr] = 2.f * (acc0[r] + acc1[r]) - sqq[m] - sqn;
    }
    __syncthreads();
    if (lane < 16) {
#pragma unroll
      for (int j2 = 0; j2 < 16; ++j2) topk_insert<KE>(kv, ki, nd[lane][j2], n0 + j2);
    }
    __syncthreads();
  }
  if (lane < 16) {
    int* op = idx + ((size_t)b * N_ + m0 + lane) * KE;
#pragma unroll
    for (int j = 0; j < KE; ++j) op[j] = ki[j];
  }
}

// --------------------------------------------------- generic WMMA GEMM -----
// Y[m][n] = sum_c X1[m][c]*W[n][c] (+ sum X2[m][c]*W[n][k1+c]) + bias[n] + add[m][n]
__global__ __launch_bounds__(32) void gemm_bias_kernel(
    const float* __restrict__ X1, int k1,
    const float* __restrict__ X2, int k2,
    const float* __restrict__ W, int ldw,
    const float* __restrict__ bias,
    const float* __restrict__ addp,
    float* __restrict__ Y, int Nn) {
  int m0 = blockIdx.x * 16, n0 = blockIdx.y * 16;
  int lane = threadIdx.x, half = lane >> 4, mr = lane & 15;
  v8f acc0 = V8F_ZERO, acc1 = V8F_ZERO;

  const float2* xr = (const float2*)(X1 + (size_t)(m0 + mr) * k1);
  const float2* wr = (const float2*)(W + (size_t)(n0 + mr) * ldw);
  int c1 = k1 >> 2;  // always even here (16 or 32 chunks)
  for (int kk = 0; kk < c1; kk += 2) {
    float2 ta0 = xr[2 * kk + half];       float2 tb0 = wr[2 * kk + half];
    float2 ta1 = xr[2 * (kk + 1) + half]; float2 tb1 = wr[2 * (kk + 1) + half];
    v2f a0; a0.x = ta0.x; a0.y = ta0.y;
    v2f b0; b0.x = tb0.x; b0.y = tb0.y;
    v2f a1; a1.x = ta1.x; a1.y = ta1.y;
    v2f b1; b1.x = tb1.x; b1.y = tb1.y;
    acc0 = wmma_f32_4(a0, b0, acc0);
    acc1 = wmma_f32_4(a1, b1, acc1);
  }
  if (X2) {
    const float2* xr2 = (const float2*)(X2 + (size_t)(m0 + mr) * k2);
    const float2* wr2 = (const float2*)(W + (size_t)(n0 + mr) * ldw + k1);
    int c2 = k2 >> 2;
    for (int kk = 0; kk < c2; kk += 2) {
      float2 ta0 = xr2[2 * kk + half];       float2 tb0 = wr2[2 * kk + half];
      float2 ta1 = xr2[2 * (kk + 1) + half]; float2 tb1 = wr2[2 * (kk + 1) + half];
      v2f a0; a0.x = ta0.x; a0.y = ta0.y;
      v2f b0; b0.x = tb0.x; b0.y = tb0.y;
      v2f a1; a1.x = ta1.x; a1.y = ta1.y;
      v2f b1; b1.x = tb1.x; b1.y = tb1.y;
      acc0 = wmma_f32_4(a0, b0, acc0);
      acc1 = wmma_f32_4(a1, b1, acc1);
    }
  }
  int col = n0 + mr;
  float bs = bias ? bias[col] : 0.f;
#pragma unroll
  for (int r = 0; r < 8; ++r) {
    int row = m0 + r + 8 * half;
    float v = acc0[r] + acc1[r] + bs;
    if (addp) v += addp[(size_t)row * Nn + col];
    Y[(size_t)row * Nn + col] = v;
  }
}

// -------------------------------------- fused pos-enc + vector attention ---
// One wave per center point. pe = relu(W1*dpos+b1)*W2^T + b2 via WMMA in LDS.
__global__ __launch_bounds__(32) void attn_kernel(
    const float* __restrict__ pos, const int* __restrict__ idx_pt,
    const float* __restrict__ qproj, const float* __restrict__ kproj,
    const float* __restrict__ vproj,
    const float* __restrict__ w1, const float* __restrict__ b1,
    const float* __restrict__ w2, const float* __restrict__ b2,
    float* __restrict__ agg) {
  __shared__ int   nb[KP];
  __shared__ float pd[KP][4];
  __shared__ float hs[KP][130];   // h rows (padded stride: bank-conflict free A loads)
  __shared__ float pes[KP][130];  // pos_enc rows
  __shared__ float qs[COUT];
  __shared__ float lg[KP * NH];
  int p = blockIdx.x;
  int b = p / N_;
  int lane = threadIdx.x, half = lane >> 4, mr = lane & 15;

  if (lane < KP) nb[lane] = idx_pt[(size_t)p * KP + lane];
  __syncthreads();
  if (lane < KP) {
    const float* pc = pos + (size_t)p * 3;
    const float* pn = pos + ((size_t)b * N_ + nb[lane]) * 3;
    pd[lane][0] = pc[0] - pn[0];
    pd[lane][1] = pc[1] - pn[1];
    pd[lane][2] = pc[2] - pn[2];
  }
  for (int i = lane; i < COUT; i += 32) qs[i] = qproj[(size_t)p * COUT + i];
  __syncthreads();
  for (int i = lane; i < KP * COUT; i += 32) {
    int k = i >> 7, o = i & 127;
    float v = w1[o * 3 + 0] * pd[k][0] + w1[o * 3 + 1] * pd[k][1] + w1[o * 3 + 2] * pd[k][2] + b1[o];
    hs[k][o] = v > 0.f ? v : 0.f;
  }
  __syncthreads();

  const float2* hr = (const float2*)&hs[mr][0];
  for (int t = 0; t < 8; ++t) {
    v8f acc0 = V8F_ZERO, acc1 = V8F_ZERO;
    const float2* wr = (const float2*)(w2 + (size_t)(t * 16 + mr) * COUT);
#pragma unroll
    for (int kk = 0; kk < 32; kk += 2) {
      float2 ta0 = hr[2 * kk + half];       float2 tb0 = wr[2 * kk + half];
      float2 ta1 = hr[2 * (kk + 1) + half]; float2 tb1 = wr[2 * (kk + 1) + half];
      v2f a0; a0.x = ta0.x; a0.y = ta0.y;
      v2f b0; b0.x = tb0.x; b0.y = tb0.y;
      v2f a1; a1.x = ta1.x; a1.y = ta1.y;
      v2f b1v; b1v.x = tb1.x; b1v.y = tb1.y;
      acc0 = wmma_f32_4(a0, b0, acc0);
      acc1 = wmma_f32_4(a1, b1v, acc1);
    }
    float bias = b2[t * 16 + mr];
#pragma unroll
    for (int r = 0; r < 8; ++r) pes[r + 8 * half][t * 16 + mr] = acc0[r] + acc1[r] + bias;
  }
  __syncthreads();

  // attention logits: mean over head-dim of (q - k + pe)
  for (int i = lane; i < KP * NH; i += 32) {
    int k = i >> 2, h = i & 3;
    const float* kr = kproj + ((size_t)b * N_ + nb[k]) * COUT + h * DPH;
    const float* qh = &qs[h * DPH];
    const float* ph = &pes[k][h * DPH];
    float s = 0.f;
#pragma unroll
    for (int d = 0; d < DPH; ++d) s += qh[d] - kr[d] + ph[d];
    lg[i] = s * (1.0f / DPH);
  }
  __syncthreads();
  if (lane < NH) {
    float mx = -3.0e38f;
    for (int k = 0; k < KP; ++k) mx = fmaxf(mx, lg[k * NH + lane]);
    float e[KP]; float s = 0.f;
#pragma unroll
    for (int k = 0; k < KP; ++k) { float ev = __expf(lg[k * NH + lane] - mx); e[k] = ev; s += ev; }
    float inv = 1.0f / s;
#pragma unroll
    for (int k = 0; k < KP; ++k) lg[k * NH + lane] = e[k] * inv;
  }
  __syncthreads();
  for (int i = lane; i < COUT; i += 32) {
    int h = i >> 5;
    float s = 0.f;
#pragma unroll
    for (int k = 0; k < KP; ++k) {
      float vv = vproj[((size_t)b * N_ + nb[k]) * COUT + i] + pes[k][i];
      s += lg[k * NH + h] * vv;
    }
    agg[(size_t)p * COUT + i] = s;
  }
}

// --------------------------------------------------- fused EdgeConv --------
// One wave handles 4 centers -> 80 rows = exactly 5 WMMA M-tiles (no padding).
// feat LDS is overwritten in place by conv output per M-tile to halve LDS.
// The unrolled nt loop exposes 8 independent WMMA accumulator chains.
__global__ __launch_bounds__(32) void edge_kernel(
    const float* __restrict__ x, const int* __restrict__ idx_e,
    const float* __restrict__ ec_w, const float* __restrict__ ec_g,
    const float* __restrict__ ec_bb, float* __restrict__ ecout) {
  __shared__ float feat[80][130];
  __shared__ float xc[4][CIN];
  __shared__ int   nbE[80];
  __shared__ float mu_s[80], rs_s[80];
  int g  = blockIdx.x;
  int p0 = g * 4;
  int b  = p0 / N_;
  int lane = threadIdx.x, half = lane >> 4, mr = lane & 15;

  for (int i = lane; i < 80; i += 32) nbE[i] = idx_e[(size_t)p0 * KE + i];
  for (int i = lane; i < 4 * CIN; i += 32) xc[i >> 6][i & 63] = x[(size_t)p0 * CIN + i];
  __syncthreads();
  for (int i = lane; i < 80 * CIN; i += 32) {
    int r = i >> 6, c = i & 63;
    int ci = r / 20;
    int j = nbE[r];
    float xn  = x[((size_t)b * N_ + j) * CIN + c];
    float xcv = xc[ci][c];
    feat[r][c]       = xn - xcv;   // neigh - center
    feat[r][CIN + c] = xcv;        // center
  }
  __syncthreads();

  for (int mt = 0; mt < 5; ++mt) {
    v8f accs[8];
    const float2* fr = (const float2*)&feat[mt * 16 + mr][0];
#pragma unroll
    for (int nt = 0; nt < 8; ++nt) {
      v8f acc = V8F_ZERO;
      const float2* wr = (const float2*)(ec_w + (size_t)(nt * 16 + mr) * (2 * CIN));
#pragma unroll
      for (int kk = 0; kk < 32; ++kk) {
        float2 ta = fr[2 * kk + half]; float2 tb = wr[2 * kk + half];
        v2f av; av.x = ta.x; av.y = ta.y;
        v2f bv2; bv2.x = tb.x; bv2.y = tb.y;
        acc = wmma_f32_4(av, bv2, acc);
      }
      accs[nt] = acc;
    }
    __syncthreads();
#pragma unroll
    for (int nt = 0; nt < 8; ++nt) {
#pragma unroll
      for (int r = 0; r < 8; ++r) feat[mt * 16 + r + 8 * half][nt * 16 + mr] = accs[nt][r];
    }
    __syncthreads();
  }

  // LayerNorm stats per row (biased variance, as jnp.var)
  for (int r = lane; r < 80; r += 32) {
    float s = 0.f, s2 = 0.f;
    for (int c = 0; c < COUT; ++c) { float v = feat[r][c]; s += v; s2 += v * v; }
    float m   = s  * (1.0f / COUT);
    float var = s2 * (1.0f / COUT) - m * m;
    mu_s[r] = m; rs_s[r] = rsqrtf(var + 1e-5f);
  }
  __syncthreads();
  // LN affine + leaky-relu + max over k
  for (int ci = 0; ci < 4; ++ci) {
    for (int o = lane; o < COUT; o += 32) {
      float gg = ec_g[o], bb2 = ec_bb[o];
      float mx = -3.0e38f;
      for (int k = 0; k < KE; ++k) {
        int r = ci * 20 + k;
        float v = (feat[r][o] - mu_s[r]) * rs_s[r] * gg + bb2;
        v = v > 0.f ? v : 0.2f * v;
        mx = fmaxf(mx, v);
      }
      ecout[(size_t)(p0 + ci) * COUT + o] = mx;
    }
  }
}

// ---------------------------------------------------------------------------
extern "C" void kernel_launch(void* const* d_in, const int* in_sizes, int n_in,
                              void* d_out, int out_size, void* d_ws, size_t ws_size,
                              hipStream_t stream) {
  (void)in_sizes; (void)n_in; (void)out_size; (void)ws_size;
  const float* x      = (const float*)d_in[0];
  const float* pos    = (const float*)d_in[1];
  const float* ec_w   = (const float*)d_in[2];
  const float* ec_g   = (const float*)d_in[3];
  const float* ec_b   = (const float*)d_in[4];
  const float* wq     = (const float*)d_in[5];
  const float* bq     = (const float*)d_in[6];
  const float* wk     = (const float*)d_in[7];
  const float* bk     = (const float*)d_in[8];
  const float* wv     = (const float*)d_in[9];
  const float* bv     = (const float*)d_in[10];
  const float* pm_w1  = (const float*)d_in[11];
  const float* pm_b1  = (const float*)d_in[12];
  const float* pm_w2  = (const float*)d_in[13];
  const float* pm_b2  = (const float*)d_in[14];
  const float* fc_w   = (const float*)d_in[15];
  const float* fc_b   = (const float*)d_in[16];
  const float* sc_w   = (const float*)d_in[17];
  const float* sc_b   = (const float*)d_in[18];
  const float* proj_w = (const float*)d_in[19];
  const float* proj_b = (const float*)d_in[20];

  const int BN = B_ * N_;
  float* ws     = (float*)d_ws;
  float* sqx    = ws;                                   // BN
  float* sqp    = sqx + BN;                             // BN
  int*   idx_pt = (int*)(sqp + BN);                     // BN*KP
  int*   idx_e  = idx_pt + (size_t)BN * KP;             // BN*KE
  float* qproj  = (float*)(idx_e + (size_t)BN * KE);    // BN*COUT
  float* kproj  = qproj + (size_t)BN * COUT;
  float* vproj  = kproj + (size_t)BN * COUT;
  float* scp    = vproj + (size_t)BN * COUT;
  float* aggb   = scp + (size_t)BN * COUT;
  float* ptout  = kproj;  // dead after attn -> reuse
  float* ecout  = vproj;  // dead after attn -> reuse

  sqnorm_kernel<<<BN / 256, 256, 0, stream>>>(x, pos, sqx, sqp);
  knn_pos_kernel<<<BN / 16, 32, 0, stream>>>(pos, sqp, idx_pt);
  knn_feat_kernel<<<BN / 16, 32, 0, stream>>>(x, sqx, idx_e);

  dim3 gproj(BN / 16, COUT / 16);
  gemm_bias_kernel<<<gproj, 32, 0, stream>>>(x, CIN, nullptr, 0, wq,   CIN, bq,   nullptr, qproj, COUT);
  gemm_bias_kernel<<<gproj, 32, 0, stream>>>(x, CIN, nullptr, 0, wk,   CIN, bk,   nullptr, kproj, COUT);
  gemm_bias_kernel<<<gproj, 32, 0, stream>>>(x, CIN, nullptr, 0, wv,   CIN, bv,   nullptr, vproj, COUT);
  gemm_bias_kernel<<<gproj, 32, 0, stream>>>(x, CIN, nullptr, 0, sc_w, CIN, sc_b, nullptr, scp,   COUT);

  attn_kernel<<<BN, 32, 0, stream>>>(pos, idx_pt, qproj, kproj, vproj,
                                     pm_w1, pm_b1, pm_w2, pm_b2, aggb);
  edge_kernel<<<BN / 4, 32, 0, stream>>>(x, idx_e, ec_w, ec_g, ec_b, ecout);

  // pt_out = fc(agg) + shortcut  (ptout aliases kproj, safe: kproj dead)
  gemm_bias_kernel<<<gproj, 32, 0, stream>>>(aggb, COUT, nullptr, 0, fc_w, COUT, fc_b, scp, ptout, COUT);
  // out = proj([pt_out, ec_out]) + proj_b
  gemm_bias_kernel<<<gproj, 32, 0, stream>>>(ptout, COUT, ecout, COUT, proj_w, 2 * COUT,
                                             proj_b, nullptr, (float*)d_out, COUT);
}